// DGI_73057393705342
// MI455X (gfx1250) — compile-verified
//
#include <hip/hip_runtime.h>
#include <hip/hip_bf16.h>

typedef __attribute__((ext_vector_type(2))) float v2f;
typedef __attribute__((ext_vector_type(8))) float v8f;

#define NH 128
#define EPS 1e-5f

// ---------------- degree kernels ----------------
__global__ void k_deg_init(float* deg, int N) {
  int i = blockIdx.x * blockDim.x + threadIdx.x;
  if (i < N) deg[i] = 1.0f;  // self loop
}

__global__ void k_deg_edge(const long long* __restrict__ ei, float* deg, int E) {
  int e = blockIdx.x * blockDim.x + threadIdx.x;
  if (e < E) atomicAdd(&deg[(int)ei[(size_t)E + e]], 1.0f);
}

__global__ void k_dinv(float* deg, int N) {
  int i = blockIdx.x * blockDim.x + threadIdx.x;
  if (i < N) deg[i] = rsqrtf(fmaxf(deg[i], 1.0f));
}

// ---------------- GEMM h = X @ W via f32 WMMA ----------------
// Block = 256 threads = 8 waves. Block computes rows [16*bid, 16*bid+16),
// wave w computes cols [16w, 16w+16). W staged transposed in LDS (pad 130).
__global__ __launch_bounds__(256) void k_gemm(const float* __restrict__ X,
                                              const float* __restrict__ W,
                                              float* __restrict__ H) {
  __shared__ float Wt[NH * 130];
  int tid = threadIdx.x;
  for (int idx = tid; idx < NH * NH; idx += 256) {
    int k = idx >> 7, n = idx & 127;
    Wt[n * 130 + k] = W[idx];
  }
  __syncthreads();

  int wave = tid >> 5;
  int lane = tid & 31;
  int lo = lane & 15;
  int hi = lane >> 4;
  int row0 = blockIdx.x * 16;
  int col = wave * 16 + lo;

  const float* xrow = X + (size_t)(row0 + lo) * NH;  // A: row M=lo
  const float* wcol = &Wt[col * 130];                // B: col N=lo (transposed)

  v8f acc = {};
#pragma unroll
  for (int k0 = 0; k0 < NH; k0 += 4) {
    int k = k0 + 2 * hi;                 // low lanes K,K+1 ; high lanes K+2,K+3
    v2f a = *(const v2f*)(xrow + k);
    v2f b = *(const v2f*)(wcol + k);
    acc = __builtin_amdgcn_wmma_f32_16x16x4_f32(false, a, false, b,
                                                (short)0, acc, false, false);
  }

  // D layout: VGPR r -> M = r + 8*hi, N = lo
  float* hbase = H + (size_t)row0 * NH + col;
#pragma unroll
  for (int r = 0; r < 8; ++r) {
    hbase[(size_t)(r + 8 * hi) * NH] = acc[r];
  }
}

// ---------------- self-loop init: agg = dinv^2 * h ----------------
__global__ void k_selfinit(const float* __restrict__ hx, const float* __restrict__ hp,
                           const float* __restrict__ dinv,
                           float* agg1, float* agg2, int total) {
  int i = blockIdx.x * blockDim.x + threadIdx.x;
  if (i >= total) return;
  int row = i >> 7;
  float w = dinv[row] * dinv[row];
  agg1[i] = w * hx[i];
  agg2[i] = w * hp[i];
}

// ---------------- edge scatter: one wave per edge, both graphs ----------------
__global__ void k_scatter(const long long* __restrict__ ei, const float* __restrict__ dinv,
                          const float* __restrict__ hx, const float* __restrict__ hp,
                          float* agg1, float* agg2, int E) {
  int g = blockIdx.x * blockDim.x + threadIdx.x;
  int e = g >> 5;
  if (e >= E) return;
  int lane = g & 31;
  int s = (int)ei[e];
  int d = (int)ei[(size_t)E + e];
  float norm = dinv[s] * dinv[d];

  float4 v1 = ((const float4*)(hx + (size_t)s * NH))[lane];
  float4 v2 = ((const float4*)(hp + (size_t)s * NH))[lane];
  float* o1 = agg1 + (size_t)d * NH + lane * 4;
  float* o2 = agg2 + (size_t)d * NH + lane * 4;
  atomicAdd(o1 + 0, norm * v1.x);
  atomicAdd(o1 + 1, norm * v1.y);
  atomicAdd(o1 + 2, norm * v1.z);
  atomicAdd(o1 + 3, norm * v1.w);
  atomicAdd(o2 + 0, norm * v2.x);
  atomicAdd(o2 + 1, norm * v2.y);
  atomicAdd(o2 + 2, norm * v2.z);
  atomicAdd(o2 + 3, norm * v2.w);
}

// ---------------- bias + PReLU (in place) ----------------
__global__ void k_bias_prelu(float* agg, const float* __restrict__ b,
                             const float* __restrict__ a_p, int total) {
  int i = blockIdx.x * blockDim.x + threadIdx.x;
  if (i >= total) return;
  float v = agg[i] + b[i & 127];
  float a = a_p[0];
  agg[i] = v >= 0.0f ? v : a * v;
}

// ---------------- zero small accumulators ----------------
__global__ void k_zero256(float* p) { p[threadIdx.x] = 0.0f; }

// ---------------- column sum / sumsq of h1 ----------------
__global__ __launch_bounds__(128) void k_stats(const float* __restrict__ h1,
                                               float* colsum, float* colsumsq, int N) {
  int c = threadIdx.x;  // channel, coalesced across threads
  float s = 0.0f, s2 = 0.0f;
  for (int i = blockIdx.x; i < N; i += gridDim.x) {
    float v = h1[(size_t)i * NH + c];
    s += v;
    s2 += v * v;
  }
  atomicAdd(&colsum[c], s);
  atomicAdd(&colsumsq[c], s2);
}

// ---------------- fold BN + summary + discriminator weights ----------------
// c = sigmoid(beta) exactly (column mean of batch-normed h1 is beta).
// wc = disc_W @ c ; wc1 = wc * scale ; const1 = disc_b + sum((beta-mu*scale)*wc)
__global__ __launch_bounds__(128) void k_small(const float* __restrict__ colsum,
                                               const float* __restrict__ colsumsq,
                                               const float* __restrict__ gamma,
                                               const float* __restrict__ beta,
                                               const float* __restrict__ discW,
                                               const float* __restrict__ discb,
                                               float* wc1, float* wc2, float* consts, int N) {
  __shared__ float cvec[NH];
  __shared__ float red[NH];
  int t = threadIdx.x;
  float invN = 1.0f / (float)N;
  float mu = colsum[t] * invN;
  float var = colsumsq[t] * invN - mu * mu;
  float scale = gamma[t] * rsqrtf(var + EPS);
  cvec[t] = 1.0f / (1.0f + __expf(-beta[t]));
  __syncthreads();
  float wc = 0.0f;
#pragma unroll 8
  for (int c2 = 0; c2 < NH; ++c2) wc += discW[t * NH + c2] * cvec[c2];
  wc2[t] = wc;
  wc1[t] = wc * scale;
  red[t] = (beta[t] - mu * scale) * wc;
  __syncthreads();
  for (int off = 64; off > 0; off >>= 1) {
    if (t < off) red[t] += red[t + off];
    __syncthreads();
  }
  if (t == 0) {
    consts[0] = red[0] + discb[0];
    consts[1] = discb[0];
  }
}

// ---------------- final dual mat-vec: one wave per row ----------------
__global__ void k_score(const float* __restrict__ h1, const float* __restrict__ h2,
                        const float* __restrict__ wc1, const float* __restrict__ wc2,
                        const float* __restrict__ consts, float* out, int N) {
  int g = blockIdx.x * blockDim.x + threadIdx.x;
  int row = g >> 5;
  if (row >= N) return;
  int lane = g & 31;
  float4 w1 = ((const float4*)wc1)[lane];
  float4 w2 = ((const float4*)wc2)[lane];
  float4 a1 = ((const float4*)(h1 + (size_t)row * NH))[lane];
  float4 a2 = ((const float4*)(h2 + (size_t)row * NH))[lane];
  float d1 = a1.x * w1.x + a1.y * w1.y + a1.z * w1.z + a1.w * w1.w;
  float d2 = a2.x * w2.x + a2.y * w2.y + a2.z * w2.z + a2.w * w2.w;
#pragma unroll
  for (int off = 16; off > 0; off >>= 1) {
    d1 += __shfl_down(d1, off, 32);
    d2 += __shfl_down(d2, off, 32);
  }
  if (lane == 0) {
    out[row] = d1 + consts[0];
    out[(size_t)N + row] = d2 + consts[1];
  }
}

extern "C" void kernel_launch(void* const* d_in, const int* in_sizes, int n_in,
                              void* d_out, int out_size, void* d_ws, size_t ws_size,
                              hipStream_t stream) {
  const float* x        = (const float*)d_in[0];
  const float* xp       = (const float*)d_in[1];
  const long long* ei   = (const long long*)d_in[2];
  const float* W        = (const float*)d_in[3];
  const float* b        = (const float*)d_in[4];
  const float* prelu_a  = (const float*)d_in[5];
  const float* bn_gamma = (const float*)d_in[6];
  const float* bn_beta  = (const float*)d_in[7];
  const float* disc_W   = (const float*)d_in[8];
  const float* disc_b   = (const float*)d_in[9];

  const int N = in_sizes[0] / NH;
  const int E = in_sizes[2] / 2;
  const int total = N * NH;

  char* ws = (char*)d_ws;
  float* hx   = (float*)ws;  ws += (size_t)total * 4;
  float* hp   = (float*)ws;  ws += (size_t)total * 4;
  float* agg1 = (float*)ws;  ws += (size_t)total * 4;
  float* agg2 = (float*)ws;  ws += (size_t)total * 4;
  float* dinv = (float*)ws;  ws += (size_t)N * 4;
  float* colsum   = (float*)ws;  ws += NH * 4;   // colsum+colsumsq contiguous
  float* colsumsq = (float*)ws;  ws += NH * 4;
  float* wc1  = (float*)ws;  ws += NH * 4;
  float* wc2  = (float*)ws;  ws += NH * 4;
  float* consts = (float*)ws;

  // 1. degrees -> dinv
  k_deg_init<<<(N + 255) / 256, 256, 0, stream>>>(dinv, N);
  k_deg_edge<<<(E + 255) / 256, 256, 0, stream>>>(ei, dinv, E);
  k_dinv<<<(N + 255) / 256, 256, 0, stream>>>(dinv, N);

  // 2. dense GEMMs via f32 WMMA (N is a multiple of 16)
  k_gemm<<<N / 16, 256, 0, stream>>>(x, W, hx);
  k_gemm<<<N / 16, 256, 0, stream>>>(xp, W, hp);

  // 3. aggregation: self loops then edge scatter
  k_selfinit<<<(total + 255) / 256, 256, 0, stream>>>(hx, hp, dinv, agg1, agg2, total);
  k_scatter<<<(int)(((size_t)E * 32 + 255) / 256), 256, 0, stream>>>(ei, dinv, hx, hp,
                                                                     agg1, agg2, E);

  // 4. bias + PReLU (in place)
  k_bias_prelu<<<(total + 255) / 256, 256, 0, stream>>>(agg1, b, prelu_a, total);
  k_bias_prelu<<<(total + 255) / 256, 256, 0, stream>>>(agg2, b, prelu_a, total);

  // 5. BN stats over h1
  k_zero256<<<1, 256, 0, stream>>>(colsum);  // zeroes colsum + colsumsq
  k_stats<<<256, 128, 0, stream>>>(agg1, colsum, colsumsq, N);

  // 6. fold BN + sigmoid(beta) summary + disc_W into two weight vectors
  k_small<<<1, 128, 0, stream>>>(colsum, colsumsq, bn_gamma, bn_beta, disc_W, disc_b,
                                 wc1, wc2, consts, N);

  // 7. final scores
  k_score<<<(int)(((size_t)N * 32 + 255) / 256), 256, 0, stream>>>(agg1, agg2, wc1, wc2,
                                                                   consts, (float*)d_out, N);
}